// HybridModel_70050916598222
// MI455X (gfx1250) — compile-verified
//
#include <hip/hip_runtime.h>
#include <math.h>

typedef __attribute__((ext_vector_type(2))) float v2f;
typedef __attribute__((ext_vector_type(8))) float v8f;

// ---------------------------------------------------------------------------
// Weight prep: transpose OIHW conv weights into wT[k'][oc] with the K axis
// re-ordered as k' = (ic*3+ky)*4 + kx, kx in 0..3 (kx==3 zero-padded so every
// WMMA K-chunk of 4 is one (ic,ky) row -> branchless B addressing).
// Also folds conv bias + BN into per-channel scale/shift.
// ---------------------------------------------------------------------------
template<int CIN, int COUT>
__global__ void prep_weights(const float* __restrict__ w,  const float* __restrict__ cb,
                             const float* __restrict__ g,  const float* __restrict__ bb,
                             const float* __restrict__ m,  const float* __restrict__ v,
                             float* __restrict__ wT, float* __restrict__ sc,
                             float* __restrict__ sh)
{
  constexpr int KP = CIN * 12;
  int tid = blockIdx.x * blockDim.x + threadIdx.x;
  int nth = gridDim.x * blockDim.x;
  for (int idx = tid; idx < KP * COUT; idx += nth) {
    int k = idx / COUT, oc = idx - k * COUT;
    int g9 = k >> 2, kx = k & 3;          // g9 = ic*3+ky
    float val = 0.0f;
    if (kx < 3) {
      int ic = g9 / 3, ky = g9 - ic * 3;
      val = w[(oc * CIN + ic) * 9 + ky * 3 + kx];
    }
    wT[idx] = val;
  }
  for (int oc = tid; oc < COUT; oc += nth) {
    float inv = g[oc] / sqrtf(v[oc] + 1e-5f);
    sc[oc] = inv;
    sh[oc] = cb[oc] * inv + bb[oc] - m[oc] * inv;
  }
}

// ---------------------------------------------------------------------------
// Implicit-GEMM conv3x3 (SAME) + BN + ReLU + 2x2 maxpool on
// V_WMMA_F32_16X16X4_F32.  One wave handles (img, pooled-row, x-tile) and ALL
// output channels (OCT accumulator pairs).  Input tile staged in LDS once
// per wave (CIN x 4 rows x 19 cols, zero-padded, stride 20) -> the K-loop is
// branchless: B-fragments from ds_load, A-fragments from pre-transposed wT.
//
// Fragment layouts (CDNA5 ISA 7.12.2, fp32 16x4 A / 4x16 B / 16x16 C):
//   A/B: lane&15 = M(=oc) / N(=pixel); lanes>=16 carry K+2/K+3
//   C/D: VGPR r -> M = r (lanes 0-15) / r+8 (lanes 16-31), N = lane&15
// ---------------------------------------------------------------------------
template<int CIN, int COUT, int H, int W, int NW>
__global__ __launch_bounds__(NW * 32) void conv_bn_relu_pool(
    const float* __restrict__ in, const float* __restrict__ wT,
    const float* __restrict__ sc, const float* __restrict__ sh,
    float* __restrict__ out)
{
  constexpr int XT   = W / 16;        // x tiles of 16 conv pixels
  constexpr int OYT  = H / 2;         // pooled rows
  constexpr int OCT  = COUT / 16;     // oc tiles per wave
  constexpr int OW   = W / 2;
  constexpr int TROW = 20;            // padded tile row stride (19 used)
  constexpr int TSZ  = CIN * 4 * TROW;

  __shared__ float tile[NW * TSZ];

  const int lane = threadIdx.x & 31;
  const int wib  = threadIdx.x >> 5;
  float* tw = tile + wib * TSZ;

  constexpr int PERIMG = OYT * XT;
  int wave = blockIdx.x * NW + wib;
  int img  = wave / PERIMG;  int rem = wave - img * PERIMG;
  int oy   = rem / XT;       int xt  = rem - oy * XT;

  const float* inb  = in + (size_t)img * (CIN * H * W);
  const int    xb   = xt * 16 - 1;    // tile col c -> global x = xb + c
  const int    yb   = oy * 2 - 1;     // tile row r -> global y = yb + r

  // ---- stage zero-padded input tile (per-wave; same-wave DS order suffices)
  for (int idx = lane; idx < TSZ; idx += 32) {
    int ic  = idx / (4 * TROW);  int r = idx - ic * 4 * TROW;
    int row = r / TROW;          int c = r - row * TROW;
    int yy = yb + row, xx = xb + c;
    float vv = 0.0f;
    if (c < 19 && (unsigned)yy < (unsigned)H && (unsigned)xx < (unsigned)W)
      vv = inb[(ic * H + yy) * W + xx];
    tw[idx] = vv;
  }

  const int n    = lane & 15;
  const int hi   = lane >> 4;
  const int koff = hi * 2;

  v8f acc0[OCT] = {};   // conv row y0 = 2*oy
  v8f acc1[OCT] = {};   // conv row y1 = 2*oy+1

  for (int ic = 0; ic < CIN; ++ic) {
    const float* trow = tw + ic * 4 * TROW;
    #pragma unroll
    for (int ky = 0; ky < 3; ++ky) {
      // B fragments: pixel n, kx = koff/koff+1 (kx==3 reads in-bounds junk, A==0)
      v2f b0; b0.x = trow[ ky      * TROW + n + koff];
              b0.y = trow[ ky      * TROW + n + koff + 1];
      v2f b1; b1.x = trow[(ky + 1) * TROW + n + koff];
              b1.y = trow[(ky + 1) * TROW + n + koff + 1];
      const float* wr = wT + ((ic * 3 + ky) * 4 + koff) * COUT + n;
      #pragma unroll
      for (int t = 0; t < OCT; ++t) {
        v2f a; a.x = wr[t * 16]; a.y = wr[COUT + t * 16];
        acc0[t] = __builtin_amdgcn_wmma_f32_16x16x4_f32(false, a, false, b0, (short)0, acc0[t], false, false);
        acc1[t] = __builtin_amdgcn_wmma_f32_16x16x4_f32(false, a, false, b1, (short)0, acc1[t], false, false);
      }
    }
  }

  // ---- fused BN + ReLU + 2x2 maxpool (horizontal shfl + vertical pair)
  float* ob = out + (size_t)img * COUT * OYT * OW;
  #pragma unroll
  for (int t = 0; t < OCT; ++t) {
    #pragma unroll
    for (int r = 0; r < 8; ++r) {
      int oc = t * 16 + r + hi * 8;
      float s = sc[oc], b = sh[oc];
      float a0 = fmaxf(fmaf(acc0[t][r], s, b), 0.0f);
      float a1 = fmaxf(fmaf(acc1[t][r], s, b), 0.0f);
      float vm = fmaxf(a0, a1);
      vm = fmaxf(vm, __shfl_xor(vm, 1, 32));
      if ((n & 1) == 0) {
        int ox = (xt * 16 + n) >> 1;
        ob[((size_t)oc * OYT + oy) * OW + ox] = vm;
      }
    }
  }
}

// ---------------------------------------------------------------------------
// Fused projection (256->8) + theta=0.5*tanh + 8-qubit statevector sim +
// all 36 Z / ZZ expectations.  One wave32 per sample (8192 samples).
// Amplitude index i = lane*8 + t  (bits[7:3]=lane, bits[2:0]=t).
// Qubit q acts on index bit (7-q).
// ---------------------------------------------------------------------------
__global__ __launch_bounds__(256) void quantum_map_kernel(
    const float* __restrict__ feats,   // (512, 64, 8, 8) == (512, 16, 256)
    const float* __restrict__ pw,      // (16, 256, 8)
    const float* __restrict__ pb,      // (16, 8)
    float* __restrict__ qout)          // (8192, 36)
{
  const int lane = threadIdx.x & 31;
  const int s    = blockIdx.x * 8 + (threadIdx.x >> 5);
  const int b    = s >> 4;
  const int kb   = s & 15;

  // ---- projection z[d] = sum_c feats[b,kb,c] * pw[kb,c,d] ----
  const float* f  = feats + (size_t)b * 4096 + kb * 256;
  const float* wk = pw + kb * 2048;
  float z[8] = {0.f,0.f,0.f,0.f,0.f,0.f,0.f,0.f};
  #pragma unroll
  for (int t = 0; t < 8; ++t) {
    int c = lane + 32 * t;
    float fv = f[c];
    const float* wc = wk + c * 8;
    #pragma unroll
    for (int d = 0; d < 8; ++d) z[d] = fmaf(fv, wc[d], z[d]);
  }
  #pragma unroll
  for (int m = 16; m >= 1; m >>= 1) {
    #pragma unroll
    for (int d = 0; d < 8; ++d) z[d] += __shfl_xor(z[d], m, 32);
  }
  float cq[8], sq[8];
  #pragma unroll
  for (int d = 0; d < 8; ++d) {
    float th   = 0.5f * tanhf(z[d] + pb[kb * 8 + d]);   // ALPHA * tanh
    float half = 0.5f * th;
    cq[d] = cosf(half);
    sq[d] = sinf(half);
  }

  // ---- statevector |0..0> ----
  float v[8];
  #pragma unroll
  for (int t = 0; t < 8; ++t) v[t] = 0.0f;
  if (lane == 0) v[0] = 1.0f;

  // ---- RY(q), q=0..4: cross-lane butterflies (index bit 7-q = lane bit 4-q)
  #pragma unroll
  for (int q = 0; q < 5; ++q) {
    int msk  = 1 << (4 - q);
    float ss = (lane & msk) ? sq[q] : -sq[q];
    #pragma unroll
    for (int t = 0; t < 8; ++t) {
      float other = __shfl_xor(v[t], msk, 32);
      v[t] = fmaf(cq[q], v[t], ss * other);
    }
  }
  // q=5 (t bit2), q=6 (t bit1), q=7 (t bit0): in-register pairs
  #pragma unroll
  for (int t = 0; t < 4; ++t) {
    float a = v[t], c = v[t + 4];
    v[t]     = cq[5] * a - sq[5] * c;
    v[t + 4] = sq[5] * a + cq[5] * c;
  }
  #pragma unroll
  for (int base = 0; base < 8; base += 4)
    #pragma unroll
    for (int t = base; t < base + 2; ++t) {
      float a = v[t], c = v[t + 2];
      v[t]     = cq[6] * a - sq[6] * c;
      v[t + 2] = sq[6] * a + cq[6] * c;
    }
  #pragma unroll
  for (int t = 0; t < 8; t += 2) {
    float a = v[t], c = v[t + 1];
    v[t]     = cq[7] * a - sq[7] * c;
    v[t + 1] = sq[7] * a + cq[7] * c;
  }

  // ---- CNOT chain i=0..6: control bit(7-i), target bit(6-i) ----
  #pragma unroll
  for (int i = 0; i < 4; ++i) {              // both bits in lane portion
    int cm = 1 << (4 - i), tm = 1 << (3 - i);
    bool ctl = (lane & cm) != 0;
    #pragma unroll
    for (int t = 0; t < 8; ++t) {
      float other = __shfl_xor(v[t], tm, 32);
      if (ctl) v[t] = other;
    }
  }
  if (lane & 1) {                            // i=4: control lane bit0, target t bit2
    #pragma unroll
    for (int t = 0; t < 4; ++t) { float tmp = v[t]; v[t] = v[t + 4]; v[t + 4] = tmp; }
  }
  { float tmp = v[4]; v[4] = v[6]; v[6] = tmp;   // i=5: control t bit2, target t bit1
    tmp = v[5]; v[5] = v[7]; v[7] = tmp; }
  { float tmp = v[2]; v[2] = v[3]; v[3] = tmp;   // i=6: control t bit1, target t bit0
    tmp = v[6]; v[6] = v[7]; v[7] = tmp; }

  // ---- probabilities + in-register WHT over t (3 bits) ----
  float A[8];
  #pragma unroll
  for (int t = 0; t < 8; ++t) A[t] = v[t] * v[t];
  #pragma unroll
  for (int h = 4; h >= 1; h >>= 1)
    #pragma unroll
    for (int base = 0; base < 8; base += 2 * h)
      #pragma unroll
      for (int t = base; t < base + h; ++t) {
        float a = A[t], c = A[t + h];
        A[t] = a + c;  A[t + h] = a - c;
      }
  // ---- signed cross-lane WHT over 5 lane bits ----
  #pragma unroll
  for (int m = 1; m < 32; m <<= 1) {
    bool hb = (lane & m) != 0;
    #pragma unroll
    for (int t = 0; t < 8; ++t) {
      float other = __shfl_xor(A[t], m, 32);
      A[t] = hb ? (other - A[t]) : (A[t] + other);
    }
  }

  // ---- emit 36 observables: qubit q -> lanemask L[q], tmask T[q] ----
  constexpr int kLm[36] = {16,8,4,2,1,0,0,0,
                           24,20,18,17,16,16,16,
                           12,10,9,8,8,8,
                           6,5,4,4,4,
                           3,2,2,2,
                           1,1,1,
                           0,0,0};
  constexpr int kTm[36] = {0,0,0,0,0,4,2,1,
                           0,0,0,0,4,2,1,
                           0,0,0,4,2,1,
                           0,0,4,2,1,
                           0,4,2,1,
                           4,2,1,
                           6,5,3};
  float* o = qout + (size_t)s * 36;
  #pragma unroll
  for (int j = 0; j < 36; ++j)
    if (lane == kLm[j]) o[j] = A[kTm[j]];
}

// ---------------------------------------------------------------------------
// Classifier: (512,576) @ (10,576)^T + bias  (trivial; 3M MACs)
// ---------------------------------------------------------------------------
__global__ __launch_bounds__(256) void classifier_kernel(
    const float* __restrict__ q, const float* __restrict__ w,
    const float* __restrict__ bias, float* __restrict__ out)
{
  int idx = blockIdx.x * blockDim.x + threadIdx.x;
  if (idx >= 512 * 10) return;
  int b = idx / 10, o = idx - b * 10;
  const float* qr = q + (size_t)b * 576;
  const float* wr = w + (size_t)o * 576;
  float acc = bias[o];
  for (int f = 0; f < 576; ++f) acc = fmaf(qr[f], wr[f], acc);
  out[idx] = acc;
}

// ---------------------------------------------------------------------------
extern "C" void kernel_launch(void* const* d_in, const int* in_sizes, int n_in,
                              void* d_out, int out_size, void* d_ws, size_t ws_size,
                              hipStream_t stream)
{
  const float* x    = (const float*)d_in[0];
  const float* w1   = (const float*)d_in[1];
  const float* cb1  = (const float*)d_in[2];
  const float* g1   = (const float*)d_in[3];
  const float* bb1  = (const float*)d_in[4];
  const float* m1   = (const float*)d_in[5];
  const float* vv1  = (const float*)d_in[6];
  const float* w2   = (const float*)d_in[7];
  const float* cb2  = (const float*)d_in[8];
  const float* g2   = (const float*)d_in[9];
  const float* bb2  = (const float*)d_in[10];
  const float* m2   = (const float*)d_in[11];
  const float* vv2  = (const float*)d_in[12];
  const float* w3   = (const float*)d_in[13];
  const float* cb3  = (const float*)d_in[14];
  const float* g3   = (const float*)d_in[15];
  const float* bb3  = (const float*)d_in[16];
  const float* m3   = (const float*)d_in[17];
  const float* vv3  = (const float*)d_in[18];
  const float* pw   = (const float*)d_in[19];
  const float* pb   = (const float*)d_in[20];
  const float* cw   = (const float*)d_in[21];
  const float* cbs  = (const float*)d_in[22];
  float* out = (float*)d_out;

  // workspace layout (~60 MB activations + ~124 KB prepped weights)
  float* o1  = (float*)d_ws;                        // 512*16*32*32
  float* o2  = o1  + (size_t)512 * 16 * 32 * 32;    // 512*32*16*16
  float* o3  = o2  + (size_t)512 * 32 * 16 * 16;    // 512*64*8*8
  float* q   = o3  + (size_t)512 * 64 * 8 * 8;      // 8192*36
  float* wt1 = q   + (size_t)8192 * 36;             // 36*16
  float* sc1 = wt1 + 36 * 16;   float* sh1 = sc1 + 16;
  float* wt2 = sh1 + 16;                            // 192*32
  float* sc2 = wt2 + 192 * 32;  float* sh2 = sc2 + 32;
  float* wt3 = sh2 + 32;                            // 384*64
  float* sc3 = wt3 + 384 * 64;  float* sh3 = sc3 + 64;

  prep_weights< 3, 16><<<2, 256, 0, stream>>>(w1, cb1, g1, bb1, m1, vv1, wt1, sc1, sh1);
  prep_weights<16, 32><<<8, 256, 0, stream>>>(w2, cb2, g2, bb2, m2, vv2, wt2, sc2, sh2);
  prep_weights<32, 64><<<16, 256, 0, stream>>>(w3, cb3, g3, bb3, m3, vv3, wt3, sc3, sh3);

  // waves: 512*32*4=65536 (8/blk); 512*16*2=16384 (8/blk); 512*8*1=4096 (4/blk)
  conv_bn_relu_pool< 3, 16, 64, 64, 8><<<8192, 256, 0, stream>>>(x,  wt1, sc1, sh1, o1);
  conv_bn_relu_pool<16, 32, 32, 32, 8><<<2048, 256, 0, stream>>>(o1, wt2, sc2, sh2, o2);
  conv_bn_relu_pool<32, 64, 16, 16, 4><<<1024, 128, 0, stream>>>(o2, wt3, sc3, sh3, o3);
  quantum_map_kernel<<<1024, 256, 0, stream>>>(o3, pw, pb, q);
  classifier_kernel<<<20, 256, 0, stream>>>(q, cw, cbs, out);
}